// set_warping_loss_endflow_forward_5463198401053
// MI455X (gfx1250) — compile-verified
//
#include <hip/hip_runtime.h>
#include <hip/hip_bf16.h>

typedef __attribute__((ext_vector_type(2))) float v2f;
typedef __attribute__((ext_vector_type(8))) float v8f;

namespace {
constexpr int Bsz = 4;
constexpr int Csz = 3;
constexpr int Hsz = 1024;
constexpr int Wsz = 1024;
constexpr int HW  = Hsz * Wsz;                 // 1,048,576
constexpr int NPIX = Bsz * HW;                 // 4,194,304 (pixels per splat)
constexpr int NTOT = Bsz * Csz * HW;           // 12,582,912 (accumulator elems)
constexpr size_t ACC_BYTES = (size_t)NTOT * sizeof(float);
}

// ---------------------------------------------------------------------------
// Bilinear forward-splat (scatter-add) of one image with scaled flow.
// mode==0: scale = 1/t   ; mode==1: scale = 1/(1-t)
// One thread per source pixel, 3 channels, up to 12 hw fp32 atomics.
// ---------------------------------------------------------------------------
__global__ void splat_kernel(const float* __restrict__ im,
                             const float* __restrict__ flow,
                             const float* __restrict__ tvals,
                             float* __restrict__ acc,
                             int mode) {
    int i = blockIdx.x * blockDim.x + threadIdx.x;
    if (i >= NPIX) return;

    int b = i >> 20;            // / HW
    int p = i & (HW - 1);       // % HW
    int y = p >> 10;            // / W
    int x = p & (Wsz - 1);      // % W

    float t = tvals[b];
    float scale = mode ? (1.0f / (1.0f - t)) : (1.0f / t);

    const float* fb = flow + (size_t)b * 2 * HW;
    float X = (float)x + scale * fb[p];
    float Y = (float)y + scale * fb[HW + p];

    float x0f = floorf(X), y0f = floorf(Y);
    float fx = X - x0f, fy = Y - y0f;
    int ix0 = (int)x0f, iy0 = (int)y0f;
    int ix1 = ix0 + 1,  iy1 = iy0 + 1;

    float wx0 = 1.0f - fx, wy0 = 1.0f - fy;
    float w00 = wx0 * wy0, w10 = fx * wy0;
    float w01 = wx0 * fy,  w11 = fx * fy;

    bool vx0 = (ix0 >= 0) & (ix0 < Wsz);
    bool vx1 = (ix1 >= 0) & (ix1 < Wsz);
    bool vy0 = (iy0 >= 0) & (iy0 < Hsz);
    bool vy1 = (iy1 >= 0) & (iy1 < Hsz);

    const float* imb  = im  + (size_t)b * Csz * HW;
    float*       accb = acc + (size_t)b * Csz * HW;
    int o00 = iy0 * Wsz + ix0;
    int o10 = iy0 * Wsz + ix1;
    int o01 = iy1 * Wsz + ix0;
    int o11 = iy1 * Wsz + ix1;

#pragma unroll
    for (int c = 0; c < Csz; ++c) {
        float v = imb[c * HW + p];
        float* ac = accb + c * HW;
        if (vx0 & vy0) unsafeAtomicAdd(ac + o00, v * w00);  // global_atomic_add_f32
        if (vx1 & vy0) unsafeAtomicAdd(ac + o10, v * w10);
        if (vx0 & vy1) unsafeAtomicAdd(ac + o01, v * w01);
        if (vx1 & vy1) unsafeAtomicAdd(ac + o11, v * w11);
    }
}

// ---------------------------------------------------------------------------
// sum |a - t| over n4 float4's. Per-lane grid-stride partials, then a wave32
// reduction through V_WMMA_F32_16X16X4_F32 (B = ones): A row m is
// [p_m, 0, p_{m+16}, 0] so D[m][n] = p_m + p_{m+16}; the 8 D VGPRs per lane
// hold 8 distinct rows -> 8 VALU adds + shfl_xor(16) yields the wave total.
// Launch only full 256-thread blocks; no early return (EXEC must be all 1s).
// ---------------------------------------------------------------------------
__global__ void reduce_absdiff_kernel(const float4* __restrict__ a,
                                      const float4* __restrict__ tgt,
                                      float* __restrict__ sum,
                                      int n4) {
    float partial = 0.0f;
    int stride = gridDim.x * blockDim.x;
    for (int i = blockIdx.x * blockDim.x + threadIdx.x; i < n4; i += stride) {
        __builtin_prefetch(&a[i + stride], 0, 1);      // global_prefetch_b8
        float4 av = a[i];
        float4 tv = tgt[i];
        partial += fabsf(av.x - tv.x) + fabsf(av.y - tv.y)
                 + fabsf(av.z - tv.z) + fabsf(av.w - tv.w);
    }

    v2f A;  A[0] = partial; A[1] = 0.0f;
    v2f Bm; Bm[0] = 1.0f;   Bm[1] = 1.0f;
    v8f Cm = {};
    // 8 args: (neg_a, A, neg_b, B, c_mod, C, reuse_a, reuse_b)
    v8f D = __builtin_amdgcn_wmma_f32_16x16x4_f32(
        false, A, false, Bm, (short)0, Cm, false, false);

    float s = D[0] + D[1] + D[2] + D[3] + D[4] + D[5] + D[6] + D[7];
    s += __shfl_xor(s, 16, 32);          // combine the two half-wave row groups
    if ((threadIdx.x & 31) == 0) {
        unsafeAtomicAdd(sum, s);
    }
}

__global__ void finalize_kernel(const float* __restrict__ sums,
                                float* __restrict__ out) {
    if (threadIdx.x == 0 && blockIdx.x == 0) {
        out[0] = (sums[0] + sums[1]) * (1.0f / (float)NTOT);
    }
}

// ---------------------------------------------------------------------------
extern "C" void kernel_launch(void* const* d_in, const int* in_sizes, int n_in,
                              void* d_out, int out_size, void* d_ws, size_t ws_size,
                              hipStream_t stream) {
    const float* flows = (const float*)d_in[0];   // [2,B,2,H,W]
    const float* im0   = (const float*)d_in[1];   // [B,C,H,W]
    const float* im1   = (const float*)d_in[2];   // [B,C,H,W]
    const float* tval  = (const float*)d_in[3];   // [B]
    float* out = (float*)d_out;

    float* acc  = (float*)d_ws;                               // NTOT floats
    float* sums = (float*)((char*)d_ws + ACC_BYTES);          // 2 floats

    const float* flow0 = flows;                               // flows[0]
    const float* flow1 = flows + (size_t)Bsz * 2 * HW;        // flows[1]

    const int SPLAT_BLK = 256;
    const int SPLAT_GRID = (NPIX + SPLAT_BLK - 1) / SPLAT_BLK;
    const int RED_BLK = 256;
    const int RED_GRID = 1024;
    const int N4 = NTOT / 4;

    // zero accumulator + both loss sums (contiguous)
    hipMemsetAsync(d_ws, 0, ACC_BYTES + 2 * sizeof(float), stream);

    // warped_fir = splat(im0, flows[0]/t); s0 = sum |warped_fir - im1|
    splat_kernel<<<SPLAT_GRID, SPLAT_BLK, 0, stream>>>(im0, flow0, tval, acc, 0);
    reduce_absdiff_kernel<<<RED_GRID, RED_BLK, 0, stream>>>(
        (const float4*)acc, (const float4*)im1, sums + 0, N4);

    // reuse accumulator for the second splat
    hipMemsetAsync(acc, 0, ACC_BYTES, stream);

    // warped_sec = splat(im1, flows[1]/(1-t)); s1 = sum |warped_sec - im0|
    splat_kernel<<<SPLAT_GRID, SPLAT_BLK, 0, stream>>>(im1, flow1, tval, acc, 1);
    reduce_absdiff_kernel<<<RED_GRID, RED_BLK, 0, stream>>>(
        (const float4*)acc, (const float4*)im0, sums + 1, N4);

    finalize_kernel<<<1, 64, 0, stream>>>(sums, out);
}